// DeclarativeLayer_231928234122
// MI455X (gfx1250) — compile-verified
//
#include <hip/hip_runtime.h>
#include <math.h>

// ---------------------------------------------------------------------------
// ADMM trajectory optimizer on gfx1250: FP32 WMMA (V_WMMA_F32_16X16X4_F32),
// 16 batch elements per wave32, D-layout elementwise trig, LDS transpose
// staging for A operands, constants built on-device into d_ws.
// ---------------------------------------------------------------------------

namespace {
constexpr int kNUM = 30;
constexpr int kNVAR = 11;
constexpr int kTILE = 16;
constexpr int kMAXITER = 500;
constexpr float kRHO_EQ = 10.0f;
constexpr float kRHO_NONHOL = 1.0f;
constexpr float kRHO_PSI = 1.0f;

// d_ws float offsets
constexpr int WS_P = 0;         // 30*11
constexpr int WS_PD = 330;      // 30*11
constexpr int WS_CINVX = 660;   // 11*11
constexpr int WS_CINVP = 781;   // 11*11
constexpr int WS_GX = 902;      // 11*11  (G_x)
constexpr int WS_GPSI = 1023;   // 11*11  (G_psi)

// LDS float offsets (all even -> 8B aligned for v2f accesses)
constexpr int ST30 = 34;        // row stride for [16][30->32] staging
constexpr int ST11 = 18;        // row stride for [16][11->16] staging
constexpr int L_TMPA = 0;       // 16*34 = 544
constexpr int L_TMPB = 544;     // 544
constexpr int L_T11A = 1088;    // 16*18 = 288
constexpr int L_T11B = 1376;    // 288
constexpr int L_BPSI = 1664;    // 8 chunks * 64
constexpr int L_BPD  = 2176;    // 8 * 64
constexpr int L_CIP  = 2688;    // 3 * 64
constexpr int L_CIX  = 2880;    // 3 * 64
constexpr int L_GP   = 3072;    // 3 * 64  (-G_psi)
constexpr int L_GX   = 3264;    // 3 * 64  (-G_x)
constexpr int L_PT   = 3456;    // 2 tiles * 3 chunks * 64
constexpr int L_PDT  = 3840;    // 2 * 3 * 64
constexpr int L_PARM = 4224;    // v_init[16], psi_init[16]
constexpr int L_SIZE = 4256;
} // namespace

typedef float v8f __attribute__((ext_vector_type(8)));
typedef float v2f __attribute__((ext_vector_type(2)));

// D = A(16x4 f32) * B(4x16 f32) + C   (full-precision FP32 WMMA)
__device__ __forceinline__ v8f wmma4(v2f a, v2f b, v8f c) {
  return __builtin_amdgcn_wmma_f32_16x16x4_f32(false, a, false, b, (short)0, c,
                                               false, false);
}

// ---- layout helpers -------------------------------------------------------
// D-layout: reg r on lane l holds element (M = r + 8*(l>=16), N = l%16).
// A-layout chunk k: lane l holds (M = l%16, K = 4k + 2*(l>=16) + {0,1}).
// B-layout chunk k: lane l holds (K = 4k + 2*(l>=16) + {0,1}, N = l%16).

__device__ __forceinline__ void stage11(float* a, v8f d, int lo, int hi) {
#pragma unroll
  for (int r = 0; r < 8; ++r) a[(r + 8 * hi) * ST11 + lo] = d[r];
}

__device__ __forceinline__ void stage30(float* a, const v8f* d, int lo, int hi) {
#pragma unroll
  for (int t = 0; t < 2; ++t)
#pragma unroll
    for (int r = 0; r < 8; ++r) a[(r + 8 * hi) * ST30 + 16 * t + lo] = d[t][r];
}

__device__ __forceinline__ v2f loadA30(const float* a, int k, int lo, int hi) {
  return *(const v2f*)&a[lo * ST30 + 4 * k + 2 * hi];
}
__device__ __forceinline__ v2f loadA11(const float* a, int k, int lo, int hi) {
  return *(const v2f*)&a[lo * ST11 + 4 * k + 2 * hi];
}
__device__ __forceinline__ v2f loadB(const float* a, int k, int lane) {
  return *(const v2f*)&a[(k * 32 + lane) * 2];
}

// ---- on-device constant construction (pure compile-time math) -------------
__device__ double bern_d(int i, int nn, double t) {
  if (i < 0 || i > nn) return 0.0;
  double c = 1.0;
  for (int k = 0; k < i; ++k) c = c * (double)(nn - k) / (double)(k + 1);
  double p = 1.0, q = 1.0;
  for (int k = 0; k < i; ++k) p *= t;
  for (int k = 0; k < nn - i; ++k) q *= (1.0 - t);
  return c * p * q;
}

__device__ void inv11(const double A[kNVAR][kNVAR], double Inv[kNVAR][kNVAR]) {
  double M[kNVAR][2 * kNVAR];
  for (int r = 0; r < kNVAR; ++r)
    for (int c = 0; c < kNVAR; ++c) {
      M[r][c] = A[r][c];
      M[r][kNVAR + c] = (r == c) ? 1.0 : 0.0;
    }
  for (int col = 0; col < kNVAR; ++col) {
    int piv = col;
    double best = fabs(M[col][col]);
    for (int r = col + 1; r < kNVAR; ++r) {
      double v = fabs(M[r][col]);
      if (v > best) { best = v; piv = r; }
    }
    if (piv != col)
      for (int c = 0; c < 2 * kNVAR; ++c) {
        double t = M[col][c]; M[col][c] = M[piv][c]; M[piv][c] = t;
      }
    double s = 1.0 / M[col][col];
    for (int c = 0; c < 2 * kNVAR; ++c) M[col][c] *= s;
    for (int r = 0; r < kNVAR; ++r)
      if (r != col) {
        double f = M[r][col];
        for (int c = 0; c < 2 * kNVAR; ++c) M[r][c] -= f * M[col][c];
      }
  }
  for (int r = 0; r < kNVAR; ++r)
    for (int c = 0; c < kNVAR; ++c) Inv[r][c] = M[r][kNVAR + c];
}

__global__ void init_consts_kernel(float* __restrict__ ws) {
  if (threadIdx.x != 0 || blockIdx.x != 0) return;
  double P[kNUM][kNVAR], Pd[kNUM][kNVAR], Pdd[kNUM][kNVAR];
  const double l = 9.0;
  for (int i = 0; i < kNUM; ++i) {
    double t = (double)i / (double)(kNUM - 1);
    for (int j = 0; j < kNVAR; ++j) {
      P[i][j] = bern_d(j, 10, t);
      Pd[i][j] = 10.0 * (bern_d(j - 1, 9, t) - bern_d(j, 9, t)) / l;
      Pdd[i][j] = 90.0 *
                  (bern_d(j - 2, 8, t) - 2.0 * bern_d(j - 1, 8, t) + bern_d(j, 8, t)) /
                  (l * l);
    }
  }
  double cost_x[kNVAR][kNVAR], cost_p[kNVAR][kNVAR];
  double gx[kNVAR][kNVAR], gp[kNVAR][kNVAR];
  for (int a = 0; a < kNVAR; ++a)
    for (int b = 0; b < kNVAR; ++b) {
      double sdd = 0, sd = 0, sp = 0;
      for (int i = 0; i < kNUM; ++i) {
        sdd += Pdd[i][a] * Pdd[i][b];
        sd += Pd[i][a] * Pd[i][b];
        sp += P[i][a] * P[i][b];
      }
      double eq = P[0][a] * P[0][b] + P[kNUM - 1][a] * P[kNUM - 1][b];
      double eqp = eq + Pd[0][a] * Pd[0][b];
      gx[a][b] = 1.0 * sd + 10.0 * eq;        // rho_nonhol*Pd'Pd + rho_eq*Aeq'Aeq
      cost_x[a][b] = 1.0 * sdd + gx[a][b];    // + w_smooth*Pdd'Pdd
      gp[a][b] = 1.0 * sp + 10.0 * eqp;       // rho_psi*P'P + rho_eq*Aeqp'Aeqp
      cost_p[a][b] = 1.0 * sdd + gp[a][b];
    }
  double inv[kNVAR][kNVAR];
  inv11(cost_x, inv);
  for (int a = 0; a < kNVAR; ++a)
    for (int b = 0; b < kNVAR; ++b) ws[WS_CINVX + a * kNVAR + b] = (float)inv[a][b];
  inv11(cost_p, inv);
  for (int a = 0; a < kNVAR; ++a)
    for (int b = 0; b < kNVAR; ++b) ws[WS_CINVP + a * kNVAR + b] = (float)inv[a][b];
  for (int a = 0; a < kNVAR; ++a)
    for (int b = 0; b < kNVAR; ++b) {
      ws[WS_GX + a * kNVAR + b] = (float)gx[a][b];
      ws[WS_GPSI + a * kNVAR + b] = (float)gp[a][b];
    }
  for (int i = 0; i < kNUM; ++i)
    for (int j = 0; j < kNVAR; ++j) {
      ws[WS_P + i * kNVAR + j] = (float)P[i][j];
      ws[WS_PD + i * kNVAR + j] = (float)Pd[i][j];
    }
}

// ---- main ADMM kernel: one wave32 per 16-batch tile -----------------------
__global__ __launch_bounds__(32) void admm_kernel(const float* __restrict__ fixed,
                                                  const float* __restrict__ var,
                                                  float* __restrict__ out,
                                                  const float* __restrict__ ws) {
  __shared__ __align__(16) float lds[L_SIZE];
  const int lane = threadIdx.x;
  const int lo = lane & 15;
  const int hi = lane >> 4;
  const int base = blockIdx.x * kTILE;

  // Build B-operand-layout constants in LDS (zero-padded: pad K rows / pad N
  // cols are exact zeros, so staging-buffer garbage never contaminates GEMMs).
  auto fillB = [&](int off, int nch, auto f) {
    for (int k = 0; k < nch; ++k) {
      int K0 = 4 * k + 2 * hi;
      lds[off + (k * 32 + lane) * 2 + 0] = f(K0, lo);
      lds[off + (k * 32 + lane) * 2 + 1] = f(K0 + 1, lo);
    }
  };
  fillB(L_BPSI, 8, [&](int K, int N) {
    return (K < kNUM && N < kNVAR) ? kRHO_PSI * ws[WS_P + K * kNVAR + N] : 0.0f;
  });
  fillB(L_BPD, 8, [&](int K, int N) {
    return (K < kNUM && N < kNVAR) ? kRHO_NONHOL * ws[WS_PD + K * kNVAR + N] : 0.0f;
  });
  fillB(L_CIP, 3, [&](int K, int N) {
    return (K < kNVAR && N < kNVAR) ? ws[WS_CINVP + K * kNVAR + N] : 0.0f;
  });
  fillB(L_CIX, 3, [&](int K, int N) {
    return (K < kNVAR && N < kNVAR) ? ws[WS_CINVX + K * kNVAR + N] : 0.0f;
  });
  fillB(L_GP, 3, [&](int K, int N) {   // negated so WMMA accumulate == subtract
    return (K < kNVAR && N < kNVAR) ? -ws[WS_GPSI + K * kNVAR + N] : 0.0f;
  });
  fillB(L_GX, 3, [&](int K, int N) {
    return (K < kNVAR && N < kNVAR) ? -ws[WS_GX + K * kNVAR + N] : 0.0f;
  });
  for (int t = 0; t < 2; ++t)
    fillB(L_PT + t * 192, 3, [&](int K, int N) {  // B = P^T  [11 x 30]
      int Ng = N + 16 * t;
      return (K < kNVAR && Ng < kNUM) ? ws[WS_P + Ng * kNVAR + K] : 0.0f;
    });
  for (int t = 0; t < 2; ++t)
    fillB(L_PDT + t * 192, 3, [&](int K, int N) { // B = Pdot^T
      int Ng = N + 16 * t;
      return (K < kNVAR && Ng < kNUM) ? ws[WS_PD + Ng * kNVAR + K] : 0.0f;
    });
  if (lane < 16) {
    lds[L_PARM + lane] = fixed[(base + lane) * 5 + 2];       // v_init
    lds[L_PARM + 16 + lane] = fixed[(base + lane) * 5 + 3];  // psi_init
  }
  __syncthreads();

  // u_* = rho_eq * b_eq @ A_eq  (iteration-invariant), via one WMMA each.
  v8f u_x = {}, u_y = {}, u_psi = {};
  {
    float xi = 0.f, xf = 0.f, yi = 0.f, yf = 0.f, pi0 = 0.f, pd0 = 0.f, pf = 0.f;
    if (hi == 0) {
      xi = fixed[(base + lo) * 5 + 0];
      yi = fixed[(base + lo) * 5 + 1];
      pi0 = fixed[(base + lo) * 5 + 3];
      pd0 = fixed[(base + lo) * 5 + 4];
      xf = var[(base + lo) * 3 + 0];
      yf = var[(base + lo) * 3 + 1];
    } else {
      pf = var[(base + lo) * 3 + 2];  // psi_fin goes to K=2 (hi lane half)
    }
    float p0 = (lo < kNVAR) ? ws[WS_P + 0 * kNVAR + lo] : 0.0f;
    float pN = (lo < kNVAR) ? ws[WS_P + (kNUM - 1) * kNVAR + lo] : 0.0f;
    float pd = (lo < kNVAR) ? ws[WS_PD + 0 * kNVAR + lo] : 0.0f;
    // A_eq rows: K0=P[0], K1=P[29]  (K2,K3 = 0)
    v2f bEq;  bEq[0] = (hi == 0) ? kRHO_EQ * p0 : 0.0f;
              bEq[1] = (hi == 0) ? kRHO_EQ * pN : 0.0f;
    // A_eq_psi rows: K0=P[0], K1=Pd[0], K2=P[29], K3=0
    v2f bEqP; bEqP[0] = (hi == 0) ? kRHO_EQ * p0 : kRHO_EQ * pN;
              bEqP[1] = (hi == 0) ? kRHO_EQ * pd : 0.0f;
    v2f ax; ax[0] = xi; ax[1] = xf;       // K0=x_init, K1=x_fin (hi half zero)
    v2f ay; ay[0] = yi; ay[1] = yf;
    v2f ap; ap[0] = (hi == 0) ? pi0 : pf; // K0=psi_init, K2=psi_fin
    v2f ap1 = ap; ap1[1] = (hi == 0) ? pd0 : 0.0f;  // K1=psidot_init, K3=0
    u_x = wmma4(ax, bEq, u_x);
    u_y = wmma4(ay, bEq, u_y);
    u_psi = wmma4(ap1, bEqP, u_psi);
  }

  // State (all in D-layout): lane holds (b = r + 8*hi, col = lo + 16*t)
  v8f lx = {}, ly = {}, lpsi = {};
  v8f xd[2], yd[2], vv[2];
#pragma unroll
  for (int t = 0; t < 2; ++t)
#pragma unroll
    for (int r = 0; r < 8; ++r) {
      float vi = lds[L_PARM + r + 8 * hi];
      float pi = lds[L_PARM + 16 + r + 8 * hi];
      vv[t][r] = vi;
      xd[t][r] = vi * cosf(pi);
      yd[t][r] = vi * sinf(pi);
    }

  v8f cx = {}, cy = {}, cpsi = {};

  for (int it = 0; it < kMAXITER; ++it) {
    // --- psi subproblem ---
    v8f pt[2];
#pragma unroll
    for (int t = 0; t < 2; ++t)
#pragma unroll
      for (int r = 0; r < 8; ++r) pt[t][r] = atan2f(yd[t][r], xd[t][r]);
    __syncthreads();
    stage30(&lds[L_TMPA], pt, lo, hi);
    __syncthreads();
    v8f gpsi = lpsi + u_psi;
#pragma unroll
    for (int k = 0; k < 8; ++k)
      gpsi = wmma4(loadA30(&lds[L_TMPA], k, lo, hi), loadB(&lds[L_BPSI], k, lane), gpsi);
    __syncthreads();
    stage11(&lds[L_T11A], gpsi, lo, hi);
    __syncthreads();
    cpsi = v8f{};
#pragma unroll
    for (int k = 0; k < 3; ++k)
      cpsi = wmma4(loadA11(&lds[L_T11A], k, lo, hi), loadB(&lds[L_CIP], k, lane), cpsi);
    __syncthreads();
    stage11(&lds[L_T11B], cpsi, lo, hi);
    __syncthreads();
    v8f psiT[2];
#pragma unroll
    for (int t = 0; t < 2; ++t) {
      v8f acc = {};
#pragma unroll
      for (int k = 0; k < 3; ++k)
        acc = wmma4(loadA11(&lds[L_T11B], k, lo, hi), loadB(&lds[L_PT + t * 192], k, lane), acc);
      psiT[t] = acc;
    }
#pragma unroll
    for (int k = 0; k < 3; ++k)  // lpsi = g_psi - c_psi @ G_psi
      gpsi = wmma4(loadA11(&lds[L_T11B], k, lo, hi), loadB(&lds[L_GP], k, lane), gpsi);
    lpsi = gpsi;

    // --- x,y subproblem ---
    v8f cp[2], sp[2], vcv[2], vsv[2];
#pragma unroll
    for (int t = 0; t < 2; ++t)
#pragma unroll
      for (int r = 0; r < 8; ++r) {
        float s, c;
        sincosf(psiT[t][r], &s, &c);
        cp[t][r] = c;
        sp[t][r] = s;
        vcv[t][r] = vv[t][r] * c;  // uses carried (old) v
        vsv[t][r] = vv[t][r] * s;
      }
    __syncthreads();
    stage30(&lds[L_TMPA], vcv, lo, hi);
    stage30(&lds[L_TMPB], vsv, lo, hi);
    __syncthreads();
    v8f gx = lx + u_x, gy = ly + u_y;
#pragma unroll
    for (int k = 0; k < 8; ++k) {
      gx = wmma4(loadA30(&lds[L_TMPA], k, lo, hi), loadB(&lds[L_BPD], k, lane), gx);
      gy = wmma4(loadA30(&lds[L_TMPB], k, lo, hi), loadB(&lds[L_BPD], k, lane), gy);
    }
    __syncthreads();
    stage11(&lds[L_T11A], gx, lo, hi);
    stage11(&lds[L_T11B], gy, lo, hi);
    __syncthreads();
    cx = v8f{};
    cy = v8f{};
#pragma unroll
    for (int k = 0; k < 3; ++k) {
      cx = wmma4(loadA11(&lds[L_T11A], k, lo, hi), loadB(&lds[L_CIX], k, lane), cx);
      cy = wmma4(loadA11(&lds[L_T11B], k, lo, hi), loadB(&lds[L_CIX], k, lane), cy);
    }
    __syncthreads();
    stage11(&lds[L_T11A], cx, lo, hi);
    stage11(&lds[L_T11B], cy, lo, hi);
    __syncthreads();
#pragma unroll
    for (int t = 0; t < 2; ++t) {
      v8f ax = {}, ay = {};
#pragma unroll
      for (int k = 0; k < 3; ++k) {
        ax = wmma4(loadA11(&lds[L_T11A], k, lo, hi), loadB(&lds[L_PDT + t * 192], k, lane), ax);
        ay = wmma4(loadA11(&lds[L_T11B], k, lo, hi), loadB(&lds[L_PDT + t * 192], k, lane), ay);
      }
      xd[t] = ax;
      yd[t] = ay;
    }
    // new v; residual uses NEW v with current cos/sin
#pragma unroll
    for (int t = 0; t < 2; ++t)
#pragma unroll
      for (int r = 0; r < 8; ++r) {
        float x = xd[t][r], y = yd[t][r];
        float vn = sqrtf(x * x + y * y);
        vv[t][r] = vn;
        vcv[t][r] = vn * cp[t][r];
        vsv[t][r] = vn * sp[t][r];
      }
    __syncthreads();
    stage30(&lds[L_TMPA], vcv, lo, hi);
    stage30(&lds[L_TMPB], vsv, lo, hi);
    __syncthreads();
    // lx = lx + u_x + rho_nonhol*(v_new cos)@Pdot - c_x@G_x   (same for y)
    gx = lx + u_x;
    gy = ly + u_y;
#pragma unroll
    for (int k = 0; k < 8; ++k) {
      gx = wmma4(loadA30(&lds[L_TMPA], k, lo, hi), loadB(&lds[L_BPD], k, lane), gx);
      gy = wmma4(loadA30(&lds[L_TMPB], k, lo, hi), loadB(&lds[L_BPD], k, lane), gy);
    }
#pragma unroll
    for (int k = 0; k < 3; ++k) {
      gx = wmma4(loadA11(&lds[L_T11A], k, lo, hi), loadB(&lds[L_GX], k, lane), gx);
      gy = wmma4(loadA11(&lds[L_T11B], k, lo, hi), loadB(&lds[L_GX], k, lane), gy);
    }
    lx = gx;
    ly = gy;
  }

  // Output: [B, 63] = concat(c_x, c_y, c_psi, v)
  if (lo < kNVAR) {
#pragma unroll
    for (int r = 0; r < 8; ++r) {
      int b = base + r + 8 * hi;
      out[b * 63 + 0 + lo] = cx[r];
      out[b * 63 + 11 + lo] = cy[r];
      out[b * 63 + 22 + lo] = cpsi[r];
    }
  }
#pragma unroll
  for (int t = 0; t < 2; ++t) {
    int col = lo + 16 * t;
    if (col < kNUM) {
#pragma unroll
      for (int r = 0; r < 8; ++r)
        out[(base + r + 8 * hi) * 63 + 33 + col] = vv[t][r];
    }
  }
}

extern "C" void kernel_launch(void* const* d_in, const int* in_sizes, int n_in,
                              void* d_out, int out_size, void* d_ws, size_t ws_size,
                              hipStream_t stream) {
  const float* fixed = (const float*)d_in[0];
  const float* var = (const float*)d_in[1];
  float* out = (float*)d_out;
  float* ws = (float*)d_ws;

  hipLaunchKernelGGL(init_consts_kernel, dim3(1), dim3(1), 0, stream, ws);

  int B = in_sizes[0] / 5;                 // 2048
  int tiles = (B + kTILE - 1) / kTILE;     // 128 waves, 16 batch rows each
  hipLaunchKernelGGL(admm_kernel, dim3(tiles), dim3(32), 0, stream,
                     fixed, var, out, ws);
}